// GIN_16870631538847
// MI455X (gfx1250) — compile-verified
//
#include <hip/hip_runtime.h>

typedef float vf2 __attribute__((ext_vector_type(2)));
typedef float vf8 __attribute__((ext_vector_type(8)));

#define GN_NODES 100000
#define GN_HID   64
#define GN_EDGES 1600000

// ---------------------------------------------------------------- copy (init agg = x)
__global__ void gin_copy_f32(const float* __restrict__ src, float* __restrict__ dst, int n4) {
  int i = blockIdx.x * blockDim.x + threadIdx.x;
  if (i < n4) ((float4*)dst)[i] = ((const float4*)src)[i];
}

// ---------------------------------------------------------------- scatter-add: agg[dst] += feat[src]
template <int F>
__global__ void gin_scatter_add(const float* __restrict__ feat,
                                const int* __restrict__ srcIdx,
                                const int* __restrict__ dstIdx,
                                float* __restrict__ agg, int nedges) {
  int tid = blockIdx.x * blockDim.x + threadIdx.x;
  int e = tid / F;
  int f = tid % F;
  if (e < nedges) {
    int s = srcIdx[e];
    int d = dstIdx[e];
    unsafeAtomicAdd(&agg[(size_t)d * F + f], feat[(size_t)s * F + f]);
  }
}

// ---------------------------------------------------------------- fused 2-layer MLP (WMMA fp32)
// out[16 rows] = relu( relu(A @ W1 + b1) @ W2 + b2 ),  A = in[tile*16 : tile*16+16, 0:K)
// Weights staged in LDS pair-interleaved: Wp[k2][c] = {W[2k2][c], W[2k2+1][c]}
// so a B-fragment is one contiguous 8-byte LDS load straight into a VGPR pair.
template <int K>
__global__ __launch_bounds__(256) void gin_mlp_kernel(
    const float* __restrict__ in,   // [N, K]
    const float* __restrict__ W1,   // [K, 64]
    const float* __restrict__ b1,   // [64]
    const float* __restrict__ W2,   // [64, 64]
    const float* __restrict__ b2,   // [64]
    float* __restrict__ out,        // [N, 64]
    float* __restrict__ out2,       // optional duplicate destination (may be null)
    int ntiles) {
  constexpr int H    = 64;
  constexpr int TS   = 68;  // tbuf row stride (floats): conflict-free b64 reads / b32 writes
  constexpr int WSTR = 80;  // weight k2-row stride (vf2): 640B == +32 banks for upper half-wave

  __shared__ vf2 W1ps[(K / 2) * WSTR];
  __shared__ vf2 W2ps[(H / 2) * WSTR];
  __shared__ float b1s[H], b2s[H];
  __shared__ float tbuf[8][16 * TS];

  const int tid = threadIdx.x;
  for (int i = tid; i < (K / 2) * H; i += 256) {
    int k2 = i / H, cc = i % H;
    vf2 w; w.x = W1[(2 * k2) * H + cc]; w.y = W1[(2 * k2 + 1) * H + cc];
    W1ps[k2 * WSTR + cc] = w;
  }
  for (int i = tid; i < (H / 2) * H; i += 256) {
    int k2 = i / H, cc = i % H;
    vf2 w; w.x = W2[(2 * k2) * H + cc]; w.y = W2[(2 * k2 + 1) * H + cc];
    W2ps[k2 * WSTR + cc] = w;
  }
  if (tid < H) { b1s[tid] = b1[tid]; b2s[tid] = b2[tid]; }
  __syncthreads();

  const int wave = tid >> 5;        // wave32
  const int lane = tid & 31;
  const int col  = lane & 15;       // A row / B,C,D column
  const int half = lane >> 4;       // K-pair (A/B) and M-half (C/D) select
  float* tb = tbuf[wave];

  const int tile = blockIdx.x * 8 + wave;   // wave-uniform -> EXEC all-ones at every WMMA
  if (tile >= ntiles) return;

  // ---------------- stage 1: T = relu(A @ W1 + b1) ----------------
  vf8 c[4];
#pragma unroll
  for (int n = 0; n < 4; ++n) {
    float bb = b1s[n * 16 + col];
#pragma unroll
    for (int v = 0; v < 8; ++v) c[n][v] = bb;
  }

  const float* arow = in + (size_t)(tile * 16 + col) * K + half * 2;
#pragma unroll
  for (int t = 0; t < K / 4; ++t) {
    vf2 a = *(const vf2*)(arow + t * 4);            // A[row][kr..kr+1]
    const vf2* wrow = &W1ps[(t * 2 + half) * WSTR]; // k2 = kr/2
#pragma unroll
    for (int n = 0; n < 4; ++n) {
      vf2 bf = wrow[n * 16 + col];                  // {W[kr][c], W[kr+1][c]}
      c[n] = __builtin_amdgcn_wmma_f32_16x16x4_f32(false, a, false, bf,
                                                   (short)0, c[n], false, false);
    }
  }

  // relu + stage through LDS (same-wave DS ops are in-order on CDNA5)
#pragma unroll
  for (int n = 0; n < 4; ++n)
#pragma unroll
    for (int v = 0; v < 8; ++v)
      tb[(v + half * 8) * TS + n * 16 + col] = fmaxf(c[n][v], 0.0f);

  // ---------------- stage 2: O = relu(T @ W2 + b2) ----------------
#pragma unroll
  for (int n = 0; n < 4; ++n) {
    float bb = b2s[n * 16 + col];
#pragma unroll
    for (int v = 0; v < 8; ++v) c[n][v] = bb;
  }

#pragma unroll
  for (int t = 0; t < 16; ++t) {
    const int kr = t * 4 + half * 2;
    vf2 a = *(const vf2*)&tb[col * TS + kr];        // T[row][kr..kr+1]
    const vf2* wrow = &W2ps[(t * 2 + half) * WSTR];
#pragma unroll
    for (int n = 0; n < 4; ++n) {
      vf2 bf = wrow[n * 16 + col];
      c[n] = __builtin_amdgcn_wmma_f32_16x16x4_f32(false, a, false, bf,
                                                   (short)0, c[n], false, false);
    }
  }

  float* orow = out + (size_t)tile * 16 * H;
#pragma unroll
  for (int n = 0; n < 4; ++n)
#pragma unroll
    for (int v = 0; v < 8; ++v)
      orow[(v + half * 8) * H + n * 16 + col] = fmaxf(c[n][v], 0.0f);

  if (out2) {  // duplicate write (seeds next layer's aggregation buffer, saves a copy pass)
    float* orow2 = out2 + (size_t)tile * 16 * H;
#pragma unroll
    for (int n = 0; n < 4; ++n)
#pragma unroll
      for (int v = 0; v < 8; ++v)
        orow2[(v + half * 8) * H + n * 16 + col] = fmaxf(c[n][v], 0.0f);
  }
}

// ---------------------------------------------------------------- final 64 -> 3 FC
__global__ void gin_fc_kernel(const float* __restrict__ h, const float* __restrict__ W,
                              const float* __restrict__ b, float* __restrict__ out, int n) {
  int i = blockIdx.x * blockDim.x + threadIdx.x;
  if (i < n) {
    float a0 = b[0], a1 = b[1], a2 = b[2];
    const float* hr = h + (size_t)i * 64;
#pragma unroll
    for (int k = 0; k < 64; ++k) {
      float v = hr[k];
      a0 += v * W[k * 3 + 0];
      a1 += v * W[k * 3 + 1];
      a2 += v * W[k * 3 + 2];
    }
    out[(size_t)i * 3 + 0] = a0;
    out[(size_t)i * 3 + 1] = a1;
    out[(size_t)i * 3 + 2] = a2;
  }
}

// ---------------------------------------------------------------- launch
extern "C" void kernel_launch(void* const* d_in, const int* in_sizes, int n_in,
                              void* d_out, int out_size, void* d_ws, size_t ws_size,
                              hipStream_t stream) {
  (void)in_sizes; (void)n_in; (void)out_size; (void)ws_size;
  const float* x   = (const float*)d_in[0];
  const int*  eidx = (const int*)d_in[1];
  const float* W1a = (const float*)d_in[2];
  const float* b1a = (const float*)d_in[3];
  const float* W1b = (const float*)d_in[4];
  const float* b1b = (const float*)d_in[5];
  const float* W2a = (const float*)d_in[6];
  const float* b2a = (const float*)d_in[7];
  const float* W2b = (const float*)d_in[8];
  const float* b2b = (const float*)d_in[9];
  const float* Wfc = (const float*)d_in[10];
  const float* bfc = (const float*)d_in[11];
  float* out = (float*)d_out;

  const int* srcI = eidx;             // edge_index[0]
  const int* dstI = eidx + GN_EDGES;  // edge_index[1]

  float* A = (float*)d_ws;                       // N*64 floats (aggregation buffer)
  float* B = A + (size_t)GN_NODES * GN_HID;      // N*64 floats (feature buffer)

  const int ntiles = GN_NODES / 16;  // 6250, exact

  // layer 1: s1 = x + scatter(x);  h1 = relu(MLP1(s1))  -> B, and duplicated into A (s2 seed)
  {
    int n4 = GN_NODES * 32 / 4;
    gin_copy_f32<<<(n4 + 255) / 256, 256, 0, stream>>>(x, A, n4);
  }
  {
    long long th = (long long)GN_EDGES * 32;
    gin_scatter_add<32><<<(int)((th + 255) / 256), 256, 0, stream>>>(x, srcI, dstI, A, GN_EDGES);
  }
  gin_mlp_kernel<32><<<(ntiles + 7) / 8, 256, 0, stream>>>(A, W1a, b1a, W1b, b1b, B, A, ntiles);

  // layer 2: s2 = h1 + scatter(h1);  h2 = relu(MLP2(s2)) -> B
  {
    long long th = (long long)GN_EDGES * 64;
    gin_scatter_add<64><<<(int)((th + 255) / 256), 256, 0, stream>>>(B, srcI, dstI, A, GN_EDGES);
  }
  gin_mlp_kernel<64><<<(ntiles + 7) / 8, 256, 0, stream>>>(A, W2a, b2a, W2b, b2b, B, nullptr, ntiles);

  // head: out = h2 @ Wfc + bfc
  gin_fc_kernel<<<(GN_NODES + 255) / 256, 256, 0, stream>>>(B, Wfc, bfc, out, GN_NODES);
}